// InstrumentedModel_44976897524195
// MI455X (gfx1250) — compile-verified
//
#include <hip/hip_runtime.h>
#include <math.h>

// ---------------- problem constants ----------------
#define B_   2
#define L_   2048
#define D_   256
#define NS_  2
#define PP_  16
#define PPOS_ 16
#define PLTM_ 64
#define BL_  (B_*L_)
#define NBANK_ 64          // 32 main + 16 cross + 16 pos
#define CHUNKS_ 64
#define LC_  (L_/CHUNKS_)  // 32
#define KTILE_ 64
#define LDS_STRIDE_ 68     // 16x64 A tile, padded row stride (bank stride 4)
#define PI_F 3.14159265358979323846f

typedef float v2f __attribute__((ext_vector_type(2)));
typedef float v8f __attribute__((ext_vector_type(8)));

__device__ __forceinline__ float sigm(float x){ return 1.0f/(1.0f+expf(-x)); }

// ---------------- workspace layout (float offsets) ----------------
constexpr size_t OFF_PARAMS = 0;                         // 16
constexpr size_t OFF_YK  = 16;                           // BL*32
constexpr size_t OFF_YQ  = OFF_YK  + (size_t)BL_*32;
constexpr size_t OFF_V   = OFF_YQ  + (size_t)BL_*32;     // BL*D
constexpr size_t OFF_G   = OFF_V   + (size_t)BL_*D_;
constexpr size_t OFF_RM  = OFF_G   + (size_t)BL_*D_;
constexpr size_t OFF_RS  = OFF_RM  + (size_t)BL_*D_;
constexpr size_t OFF_YLK = OFF_RS  + (size_t)BL_*D_;     // BL*64
constexpr size_t OFF_KC  = OFF_YLK + (size_t)BL_*PLTM_;  // BL*32
constexpr size_t OFF_KS  = OFF_KC  + (size_t)BL_*32;
constexpr size_t OFF_QC  = OFF_KS  + (size_t)BL_*32;
constexpr size_t OFF_QS  = OFF_QC  + (size_t)BL_*32;
constexpr size_t OFF_JKC = OFF_QS  + (size_t)BL_*32;     // BL*16
constexpr size_t OFF_JKS = OFF_JKC + (size_t)BL_*16;
constexpr size_t OFF_JQC = OFF_JKS + (size_t)BL_*16;
constexpr size_t OFF_JQS = OFF_JQC + (size_t)BL_*16;
constexpr size_t OFF_LCOS= OFF_JQS + (size_t)BL_*16;     // BL*64
constexpr size_t OFF_LSIN= OFF_LCOS+ (size_t)BL_*PLTM_;
constexpr size_t OFF_WG  = OFF_LSIN+ (size_t)BL_*PLTM_;  // BL
constexpr size_t OFF_PC  = OFF_WG  + (size_t)BL_;        // L*16
constexpr size_t OFF_PS  = OFF_PC  + (size_t)L_*PPOS_;
constexpr size_t OFF_BR  = OFF_PS  + (size_t)L_*PPOS_;   // PLTM*D
constexpr size_t OFF_BI  = OFF_BR  + (size_t)PLTM_*D_;
constexpr size_t OFF_PERS= OFF_BI  + (size_t)PLTM_*D_;   // BL*D
constexpr size_t OFF_PART= OFF_PERS+ (size_t)BL_*D_;     // B*CHUNKS*NBANK*D float2

// ---------------- scalar parameter prep ----------------
__global__ void k_prep(const float* sw, const float* posw, const float* lrw,
                       const float* rsc, const float* rth, const float* ssc,
                       const float* sbi, const float* cnt, float* p){
  float s0=sw[0], s1=sw[1];
  float mx = fmaxf(s0,s1);
  float e0=expf(s0-mx), e1=expf(s1-mx);
  float inv = 1.0f/(e0+e1);
  p[0]=e0*inv; p[1]=e1*inv;
  p[2]=sigm(posw[0]);
  p[3]=sigm(lrw[0]);
  p[4]=fminf(fmaxf(rsc[0],1.0f),20.0f);
  p[5]=fminf(fmaxf(rth[0],0.1f),0.9f);
  p[6]=ssc[0]; p[7]=sbi[0];
  p[8]=1.0f/sqrtf(fmaxf(cnt[0],1.0f)*(float)PLTM_);
  p[9]=1.0f/(float)(NS_+1);
}

// ---------------- positional phasor table ----------------
__global__ void k_postab(const float* __restrict__ freqs, float* pc, float* ps){
  int i = blockIdx.x*blockDim.x + threadIdx.x;
  if (i >= L_*PPOS_) return;
  int l = i/PPOS_, pf = i%PPOS_;
  float ang = (float)l * freqs[pf] * (2.0f*PI_F);
  pc[i] = cosf(ang); ps[i] = sinf(ang);
}

// ---------------- repack LTM binding memory ----------------
__global__ void k_bind(const float* __restrict__ src, float* br, float* bi){
  int i = blockIdx.x*blockDim.x + threadIdx.x;
  if (i >= PLTM_*D_) return;
  br[i] = src[2*i]; bi[i] = src[2*i+1];
}

// ---------------- WMMA f32 GEMM with async-to-LDS A staging ----------------
// 128 threads = 4 waves; block computes a 16(M) x 64(N) region; A tile (16xKTILE)
// is staged into LDS with global_load_async_to_lds_b64 and shared by all waves.
// flags: bit0 = add bias, bit1 = accumulate existing C, bit2 = sigmoid epilogue
__global__ __launch_bounds__(128) void k_gemm(const float* __restrict__ A,
        const float* __restrict__ W, const float* __restrict__ bias,
        float* __restrict__ Cmat, int N, int K, int lda, int ldb, int ldc,
        int flags){
  __shared__ float sA[16*LDS_STRIDE_];
  int tid  = threadIdx.x;
  int w    = tid >> 5;              // wave id 0..3
  int lane = tid & 31;
  int half = lane >> 4, lm = lane & 15;
  int mt = blockIdx.x;
  int ntiles = N >> 4;
  int nt = blockIdx.y*4 + w;
  if (nt >= ntiles) nt = ntiles - 1;   // duplicate work, benign identical stores
  int m0  = mt*16;
  int col = nt*16 + lm;

  v8f c;
  #pragma unroll
  for (int v=0; v<8; ++v){
    int crow = m0 + v + 8*half;
    float init = 0.0f;
    if (flags & 2) init = Cmat[(size_t)crow*ldc + col];
    if (flags & 1) init += bias[col];
    c[v] = init;
  }

  for (int k0=0; k0<K; k0+=KTILE_){
    // ---- async stage A[m0:m0+16, k0:k0+64] -> LDS (each 32-lane group = one row)
    #pragma unroll
    for (int j=0; j<4; ++j){
      int f   = tid + j*128;          // float2 index, 0..511
      int row = f >> 5;               // 0..15
      int k2  = (f & 31) << 1;        // 0..62
      const float* gp = A + (size_t)(m0+row)*lda + k0 + k2;
      uint32_t loff = (uint32_t)(uintptr_t)(&sA[row*LDS_STRIDE_ + k2]);
      asm volatile("global_load_async_to_lds_b64 %0, %1, off"
                   :: "v"(loff), "v"(gp) : "memory");
    }
    asm volatile("s_wait_asynccnt 0" ::: "memory");
    __syncthreads();
    // ---- 16 WMMAs over the staged K tile
    #pragma unroll 4
    for (int kk=0; kk<KTILE_; kk+=4){
      v2f a, b;
      float2 af = *(const float2*)&sA[lm*LDS_STRIDE_ + kk + 2*half];
      a.x = af.x; a.y = af.y;
      b.x = W[(size_t)(k0+kk+2*half  )*ldb + col];
      b.y = W[(size_t)(k0+kk+2*half+1)*ldb + col];
      c = __builtin_amdgcn_wmma_f32_16x16x4_f32(false, a, false, b, (short)0, c,
                                                false, false);
    }
    __syncthreads();
  }

  #pragma unroll
  for (int v=0; v<8; ++v){
    int crow = m0 + v + 8*half;
    float r = c[v];
    if (flags & 4) r = sigm(r);
    Cmat[(size_t)crow*ldc + col] = r;
  }
}

// ---------------- running mean / std scan over L ----------------
__global__ void k_stats(const float* __restrict__ x, float* rm, float* rsd){
  int i = blockIdx.x*blockDim.x + threadIdx.x;   // [0, B*D)
  if (i >= B_*D_) return;
  int b = i / D_, d = i % D_;
  float sx=0.0f, sxx=0.0f;
  for (int l=0;l<L_;++l){
    size_t idx = ((size_t)(b*L_+l))*D_ + d;
    float xv = x[idx];
    sx += xv; sxx += xv*xv;
    float invn = 1.0f/(float)(l+1);
    float m = sx*invn;
    float var = sxx*invn - m*m;
    rm[idx]  = m;
    rsd[idx] = sqrtf(fmaxf(var, 1e-8f));
  }
}

// ---------------- key/query phasors + joint phasors ----------------
__global__ void k_phkq(const float* __restrict__ yk, const float* __restrict__ yq,
                       float* kc, float* ks, float* qc, float* qs,
                       float* jkc, float* jks, float* jqc, float* jqs){
  int i = blockIdx.x*blockDim.x + threadIdx.x;   // [0, BL*PP)
  if (i >= BL_*PP_) return;
  int bl = i / PP_, pp = i % PP_;
  int i0 = bl*32 + pp, i1 = i0 + 16;
  float pk0 = tanhf(yk[i0])*PI_F, pk1 = tanhf(yk[i1])*PI_F;
  kc[i0]=cosf(pk0); ks[i0]=sinf(pk0);
  kc[i1]=cosf(pk1); ks[i1]=sinf(pk1);
  jkc[i]=cosf(pk0+pk1); jks[i]=sinf(pk0+pk1);
  float pq0 = tanhf(yq[i0])*PI_F, pq1 = tanhf(yq[i1])*PI_F;
  qc[i0]=cosf(pq0); qs[i0]=sinf(pq0);
  qc[i1]=cosf(pq1); qs[i1]=sinf(pq1);
  jqc[i]=cosf(pq0+pq1); jqs[i]=sinf(pq0+pq1);
}

__global__ void k_phltm(const float* __restrict__ ylk, float* lc, float* ls){
  int i = blockIdx.x*blockDim.x + threadIdx.x;
  if (i >= BL_*PLTM_) return;
  float ph = tanhf(ylk[i])*PI_F;
  lc[i]=cosf(ph); ls[i]=sinf(ph);
}

// ---------------- write-gate scan (|km| doesn't need q: |q|=1) ----------------
__global__ __launch_bounds__(32) void k_wg(const float* __restrict__ jkc,
        const float* __restrict__ jks, const float* __restrict__ p, float* wg){
  int b = blockIdx.x;
  int lane = threadIdx.x;
  __shared__ float red[16];
  float scale=p[4], thr=p[5], ss=p[6], sb=p[7];
  float kmr=0.0f, kmi=0.0f;
  for (int l=0;l<L_;++l){
    if (lane<16) red[lane] = sqrtf(kmr*kmr + kmi*kmi);
    __syncthreads();
    if (lane==0){
      float sum=0.0f;
      #pragma unroll
      for (int k=0;k<16;++k) sum += red[k];
      float mean = sum * (1.0f/16.0f);
      float norm = mean / sqrtf(fmaxf((float)l, 1.0f));
      float surprise = 0.5f*(1.0f - tanhf(scale*(norm-thr)));
      wg[b*L_ + l] = sigm(ss*(surprise-0.5f)+sb);
    }
    __syncthreads();
    if (lane<16){
      int idx = (b*L_ + l)*16 + lane;
      kmr += jkc[idx]; kmi += jks[idx];
    }
  }
}

// ---------------- chunked scan pass 1: per-chunk bank partial sums ----------------
__global__ __launch_bounds__(256) void k_scan1(const float* __restrict__ V,
    const float* __restrict__ wg,
    const float* __restrict__ kc, const float* __restrict__ ks,
    const float* __restrict__ jkc, const float* __restrict__ jks,
    const float* __restrict__ pc, const float* __restrict__ ps, float2* part){
  int chunk = blockIdx.x, b = blockIdx.y;
  int tid = threadIdx.x; int d = tid;
  __shared__ float sc[NBANK_], ssn[NBANK_], swg;
  float accR[NBANK_], accI[NBANK_];
  #pragma unroll
  for (int k=0;k<NBANK_;++k){ accR[k]=0.0f; accI[k]=0.0f; }
  for (int s=0;s<LC_;++s){
    int l = chunk*LC_ + s; int bl = b*L_ + l;
    if (tid < 32)      { sc[tid]=kc[bl*32+tid];          ssn[tid]=ks[bl*32+tid]; }
    else if (tid < 48) { int pp=tid-32; sc[tid]=jkc[bl*16+pp]; ssn[tid]=jks[bl*16+pp]; }
    else if (tid < 64) { int pf=tid-48; sc[tid]=pc[l*16+pf];   ssn[tid]=ps[l*16+pf]; }
    else if (tid == 64){ swg = wg[bl]; }
    __syncthreads();
    float v  = V[(size_t)bl*D_ + d];
    float vg = v * swg;
    #pragma unroll
    for (int k=0;k<48;++k){ accR[k] += sc[k]*vg; accI[k] += ssn[k]*vg; }
    #pragma unroll
    for (int k=48;k<NBANK_;++k){ accR[k] += sc[k]*v; accI[k] += ssn[k]*v; }
    __syncthreads();
  }
  size_t base = ((size_t)(b*CHUNKS_ + chunk))*NBANK_;
  #pragma unroll
  for (int k=0;k<NBANK_;++k)
    part[(base + k)*D_ + d] = make_float2(accR[k], accI[k]);
}

// ---------------- pass 2: exclusive prefix over chunks (in place) ----------------
__global__ void k_scan2(float2* part){
  int i = blockIdx.x*blockDim.x + threadIdx.x;   // [0, B*NBANK*D)
  if (i >= B_*NBANK_*D_) return;
  int d = i % D_; int bank = (i/D_) % NBANK_; int b = i/(D_*NBANK_);
  float runx=0.0f, runy=0.0f;
  for (int c=0;c<CHUNKS_;++c){
    size_t idx = (((size_t)(b*CHUNKS_ + c))*NBANK_ + bank)*D_ + d;
    float2 t = part[idx];
    part[idx] = make_float2(runx, runy);
    runx += t.x; runy += t.y;
  }
}

// ---------------- pass 3: replay with carry, retrieval + full epilogue ----------------
__global__ __launch_bounds__(256) void k_scan3(const float* __restrict__ x,
    const float* __restrict__ V, const float* __restrict__ wg,
    const float* __restrict__ g, const float* __restrict__ pers,
    const float* __restrict__ kc, const float* __restrict__ ks,
    const float* __restrict__ qc, const float* __restrict__ qs,
    const float* __restrict__ jkc, const float* __restrict__ jks,
    const float* __restrict__ jqc, const float* __restrict__ jqs,
    const float* __restrict__ pc, const float* __restrict__ ps,
    const float2* __restrict__ part, const float* __restrict__ p,
    float* __restrict__ out){
  int chunk = blockIdx.x, b = blockIdx.y;
  int tid = threadIdx.x; int d = tid;
  __shared__ float sc[NBANK_], ssn[NBANK_], rc[48], rsn[48], swg;
  float w0=p[0], w1=p[1], sposw=p[2], slrw=p[3], invp=p[8], inv3=p[9];
  float accR[NBANK_], accI[NBANK_];
  size_t base = ((size_t)(b*CHUNKS_ + chunk))*NBANK_;
  #pragma unroll
  for (int k=0;k<NBANK_;++k){
    float2 t = part[(base+k)*D_ + d];
    accR[k]=t.x; accI[k]=t.y;
  }
  for (int s=0;s<LC_;++s){
    int l = chunk*LC_ + s; int bl = b*L_ + l;
    if (tid < 32){
      sc[tid]=kc[bl*32+tid]; ssn[tid]=ks[bl*32+tid];
      rc[tid]=qc[bl*32+tid]; rsn[tid]=qs[bl*32+tid];
    } else if (tid < 48){
      int pp=tid-32;
      sc[tid]=jkc[bl*16+pp]; ssn[tid]=jks[bl*16+pp];
      rc[tid]=jqc[bl*16+pp]; rsn[tid]=jqs[bl*16+pp];
    } else if (tid < 64){
      int pf=tid-48; sc[tid]=pc[l*16+pf]; ssn[tid]=ps[l*16+pf];
    } else if (tid == 64){ swg = wg[bl]; }
    __syncthreads();
    size_t idx = (size_t)bl*D_ + d;
    float v  = V[idx];
    float vg = v * swg;
    #pragma unroll
    for (int k=0;k<48;++k){ accR[k] += sc[k]*vg; accI[k] += ssn[k]*vg; }
    #pragma unroll
    for (int k=48;k<NBANK_;++k){ accR[k] += sc[k]*v; accI[k] += ssn[k]*v; }
    // Re(memory * conj(q)) = mr*qr + mi*qi  (inclusive cumsum)
    float pb0=0.0f, pb1=0.0f, cr=0.0f, pv=0.0f;
    #pragma unroll
    for (int k=0;k<16;++k)  pb0 += accR[k]*rc[k] + accI[k]*rsn[k];
    #pragma unroll
    for (int k=16;k<32;++k) pb1 += accR[k]*rc[k] + accI[k]*rsn[k];
    #pragma unroll
    for (int k=32;k<48;++k) cr  += accR[k]*rc[k] + accI[k]*rsn[k];
    #pragma unroll
    for (int k=48;k<NBANK_;++k) pv += accR[k]*sc[k] + accI[k]*ssn[k];
    float total = inv3*(w0*pb0 + w1*pb1 + cr);
    float gate  = g[idx];
    out[idx] = x[idx] + gate*total + (1.0f-gate)*(sposw*pv)
             + slrw*(pers[idx]*invp);
    __syncthreads();
  }
}

// ---------------- host side ----------------
extern "C" void kernel_launch(void* const* d_in, const int* in_sizes, int n_in,
                              void* d_out, int out_size, void* d_ws, size_t ws_size,
                              hipStream_t stream) {
  (void)in_sizes; (void)n_in; (void)out_size; (void)ws_size;
  const float* x    = (const float*)d_in[0];
  const float* Wk   = (const float*)d_in[1];
  const float* bk   = (const float*)d_in[2];
  const float* Wq   = (const float*)d_in[3];
  const float* bq   = (const float*)d_in[4];
  const float* Wv   = (const float*)d_in[5];
  const float* bv   = (const float*)d_in[6];
  const float* setw = (const float*)d_in[7];
  const float* rsc  = (const float*)d_in[8];
  const float* rth  = (const float*)d_in[9];
  const float* ssc  = (const float*)d_in[10];
  const float* sbi  = (const float*)d_in[11];
  const float* pfr  = (const float*)d_in[12];
  const float* posw = (const float*)d_in[13];
  const float* Wg   = (const float*)d_in[14];
  const float* bg   = (const float*)d_in[15];
  const float* Wlk  = (const float*)d_in[16];
  const float* blk  = (const float*)d_in[17];
  const float* bri  = (const float*)d_in[21];
  const float* cnt  = (const float*)d_in[22];
  const float* lrw  = (const float*)d_in[27];
  float* out = (float*)d_out;
  float* ws  = (float*)d_ws;

  float* P    = ws + OFF_PARAMS;
  float* YK   = ws + OFF_YK;   float* YQ  = ws + OFF_YQ;
  float* Vb   = ws + OFF_V;    float* Gb  = ws + OFF_G;
  float* RM   = ws + OFF_RM;   float* RSd = ws + OFF_RS;
  float* YLK  = ws + OFF_YLK;
  float* KC   = ws + OFF_KC;   float* KS  = ws + OFF_KS;
  float* QC   = ws + OFF_QC;   float* QS  = ws + OFF_QS;
  float* JKC  = ws + OFF_JKC;  float* JKS = ws + OFF_JKS;
  float* JQC  = ws + OFF_JQC;  float* JQS = ws + OFF_JQS;
  float* LCO  = ws + OFF_LCOS; float* LSI = ws + OFF_LSIN;
  float* WGb  = ws + OFF_WG;
  float* PC   = ws + OFF_PC;   float* PS  = ws + OFF_PS;
  float* BR   = ws + OFF_BR;   float* BI  = ws + OFF_BI;
  float* PERS = ws + OFF_PERS;
  float2* PART = (float2*)(ws + OFF_PART);

  k_prep  <<<1, 1, 0, stream>>>(setw, posw, lrw, rsc, rth, ssc, sbi, cnt, P);
  k_postab<<<(L_*PPOS_+255)/256, 256, 0, stream>>>(pfr, PC, PS);
  k_bind  <<<(PLTM_*D_+255)/256, 256, 0, stream>>>(bri, BR, BI);

  // Dense layers on the WMMA path (f32 16x16x4, async-to-LDS A staging)
  k_gemm<<<dim3(BL_/16, 1), 128, 0, stream>>>(x,  Wk, bk, YK,  32,  D_, D_, 32,  32,  1);
  k_gemm<<<dim3(BL_/16, 1), 128, 0, stream>>>(x,  Wq, bq, YQ,  32,  D_, D_, 32,  32,  1);
  k_gemm<<<dim3(BL_/16, 4), 128, 0, stream>>>(x,  Wv, bv, Vb,  D_,  D_, D_, D_,  D_,  1);
  k_gemm<<<dim3(BL_/16, 4), 128, 0, stream>>>(x,  Wg, bg, Gb,  D_,  D_, D_, D_,  D_,  1|4);

  k_stats<<<(B_*D_+255)/256, 256, 0, stream>>>(x, RM, RSd);

  // ltm_in = [x, rm, rs]: three accumulate passes over Wlk row blocks
  k_gemm<<<dim3(BL_/16, 1), 128, 0, stream>>>(x,   Wlk,           blk,     YLK, PLTM_, D_, D_, PLTM_, PLTM_, 1);
  k_gemm<<<dim3(BL_/16, 1), 128, 0, stream>>>(RM,  Wlk+256*PLTM_, nullptr, YLK, PLTM_, D_, D_, PLTM_, PLTM_, 2);
  k_gemm<<<dim3(BL_/16, 1), 128, 0, stream>>>(RSd, Wlk+512*PLTM_, nullptr, YLK, PLTM_, D_, D_, PLTM_, PLTM_, 2);

  k_phkq <<<(BL_*PP_+255)/256,   256, 0, stream>>>(YK, YQ, KC, KS, QC, QS, JKC, JKS, JQC, JQS);
  k_phltm<<<(BL_*PLTM_+255)/256, 256, 0, stream>>>(YLK, LCO, LSI);

  // LTM read = cos@bind_r + sin@bind_i  (real part of complex retrieval)
  k_gemm<<<dim3(BL_/16, 4), 128, 0, stream>>>(LCO, BR, nullptr, PERS, D_, PLTM_, PLTM_, D_, D_, 0);
  k_gemm<<<dim3(BL_/16, 4), 128, 0, stream>>>(LSI, BI, nullptr, PERS, D_, PLTM_, PLTM_, D_, D_, 2);

  k_wg<<<B_, 32, 0, stream>>>(JKC, JKS, P, WGb);

  // chunked linear scan over L (64 chunks x 32 steps), 64 complex banks/lane
  k_scan1<<<dim3(CHUNKS_, B_), 256, 0, stream>>>(Vb, WGb, KC, KS, JKC, JKS, PC, PS, PART);
  k_scan2<<<(B_*NBANK_*D_+255)/256, 256, 0, stream>>>(PART);
  k_scan3<<<dim3(CHUNKS_, B_), 256, 0, stream>>>(x, Vb, WGb, Gb, PERS,
                                                 KC, KS, QC, QS, JKC, JKS, JQC, JQS,
                                                 PC, PS, PART, P, out);
}